// Model3_46291157517074
// MI455X (gfx1250) — compile-verified
//
#include <hip/hip_runtime.h>

typedef float v2f __attribute__((ext_vector_type(2)));
typedef float v8f __attribute__((ext_vector_type(8)));

#define NB    64      // batch
#define NH    64      // height
#define NW    2048    // width
#define NE    4       // experts
#define NCO   16      // out channels
#define NK    64      // kernel width
#define NWOUT (NW - NK + 1)   // 1985
#define XPAD  2144    // 2048 row + zero pad (worst read: 1984+15+62+1 = 2062)

// ---------------------------------------------------------------------------
// Kernel 1: global mean pool -> tiny MLP -> softmax -> aggregate expert kernels
// One block per batch sample. Output: agg[b][co][k]  (64 x 16 x 64 floats in ws)
// ---------------------------------------------------------------------------
__global__ void moe_attn_agg(const float* __restrict__ x,
                             const float* __restrict__ weight,
                             const float* __restrict__ w1,
                             const float* __restrict__ b1,
                             const float* __restrict__ w2,
                             const float* __restrict__ b2,
                             float* __restrict__ agg) {
    __shared__ float red[256];
    __shared__ float attn[NE];

    const int b   = blockIdx.x;
    const int tid = threadIdx.x;
    const float* xb = x + (size_t)b * NH * NW;

    // mean over H*W = 131072 elements, 512 per thread, coalesced
    float s = 0.0f;
    for (int i = tid; i < NH * NW; i += 256) s += xb[i];
    red[tid] = s;
    __syncthreads();
    for (int off = 128; off > 0; off >>= 1) {
        if (tid < off) red[tid] += red[tid + off];
        __syncthreads();
    }

    if (tid == 0) {
        const float pooled = red[0] * (1.0f / (float)(NH * NW));
        // Linear(1->4) + ReLU
        float h[4];
        #pragma unroll
        for (int j = 0; j < 4; ++j) {
            float v = pooled * w1[j] + b1[j];
            h[j] = v > 0.0f ? v : 0.0f;
        }
        // Linear(4->4) + softmax
        float lg[NE], mx = -3.0e38f;
        #pragma unroll
        for (int e = 0; e < NE; ++e) {
            float v = b2[e];
            #pragma unroll
            for (int j = 0; j < 4; ++j) v += h[j] * w2[j * NE + e];
            lg[e] = v;
            mx = v > mx ? v : mx;
        }
        float den = 0.0f;
        #pragma unroll
        for (int e = 0; e < NE; ++e) { lg[e] = __expf(lg[e] - mx); den += lg[e]; }
        const float inv = 1.0f / den;
        #pragma unroll
        for (int e = 0; e < NE; ++e) attn[e] = lg[e] * inv;
    }
    __syncthreads();

    // agg[co][k] = sum_e attn[e] * weight[e][co][k]   (1024 values)
    const float a0 = attn[0], a1 = attn[1], a2 = attn[2], a3 = attn[3];
    for (int i = tid; i < NCO * NK; i += 256) {
        float v = a0 * weight[i]
                + a1 * weight[1 * NCO * NK + i]
                + a2 * weight[2 * NCO * NK + i]
                + a3 * weight[3 * NCO * NK + i];
        agg[b * (NCO * NK) + i] = v;
    }
}

// ---------------------------------------------------------------------------
// Kernel 2: per-row Toeplitz GEMM via V_WMMA_F32_16X16X4_F32.
// One block per (b, h) row: 8 waves, each wave owns 16-wide output tiles.
//   D(16co x 16w) += A(16co x 4k) * B(4k x 16w),  16 K-steps cover K=64.
// A layout (f32 16x4): lane = M(co); VGPR0/1 = K={0,1} (lanes 0-15) or K={2,3}
// (lanes 16-31).  B layout (f32 4x16): lane%16 = N(w); same K split per half.
// C/D layout: VGPR r -> co = r + 8*(lane>=16), n = lane%16.
// Output (520 MB, write-once) is stored non-temporal so L2 (192 MB) stays
// resident with x (33.5 MB, re-read after the pool pass) and agg.
// ---------------------------------------------------------------------------
__global__ void moe_conv_wmma(const float* __restrict__ x,
                              const float* __restrict__ agg,
                              float* __restrict__ out) {
    __shared__ float xrow[XPAD];
    __shared__ float aw[NCO * NK];

    const int blk = blockIdx.x;
    const int b   = blk >> 6;     // / NH
    const int h   = blk & 63;     // % NH
    const int tid = threadIdx.x;

    // stage x row (8 KB) + zero pad into LDS
    const float* xsrc = x + ((size_t)b * NH + h) * NW;
    for (int i = tid; i < NW; i += 256)        xrow[i] = xsrc[i];
    for (int i = NW + tid; i < XPAD; i += 256) xrow[i] = 0.0f;
    // stage aggregated weights (4 KB)
    const float* asrc = agg + b * (NCO * NK);
    for (int i = tid; i < NCO * NK; i += 256)  aw[i] = asrc[i];
    __syncthreads();

    const int lane = tid & 31;
    const int wid  = tid >> 5;
    const int n    = lane & 15;   // N index (also M index for A loads)
    const int half = lane >> 4;

    // preload all 16 A K-slices (agg weights are tile-invariant)
    v2f a[16];
    #pragma unroll
    for (int s = 0; s < 16; ++s) {
        const int k = 4 * s + 2 * half;
        a[s].x = aw[n * NK + k];
        a[s].y = aw[n * NK + k + 1];
    }

    // per-lane output base: co = 8*half (+r below), column w0+n added per tile
    float* obase = out + (((size_t)b * NCO + 8 * half) * NH + h) * NWOUT + n;

    const int numTiles = (NWOUT + 15) / 16;   // 125 (last tile: 1 valid col)
    for (int t = wid; t < numTiles; t += 8) { // wave-uniform: EXEC all-1s
        const int w0 = t * 16;
        v8f c = {};
        const float* bp = &xrow[w0 + n + 2 * half];
        #pragma unroll
        for (int s = 0; s < 16; ++s) {
            v2f bv;
            bv.x = bp[4 * s];
            bv.y = bp[4 * s + 1];
            c = __builtin_amdgcn_wmma_f32_16x16x4_f32(
                    false, a[s], false, bv, (short)0, c, false, false);
        }
        if (w0 + n < NWOUT) {                  // predicate stores only
            float* op = obase + w0;
            #pragma unroll
            for (int r = 0; r < 8; ++r)
                __builtin_nontemporal_store(c[r], op + (size_t)r * NH * NWOUT);
        }
    }
}

// ---------------------------------------------------------------------------
extern "C" void kernel_launch(void* const* d_in, const int* in_sizes, int n_in,
                              void* d_out, int out_size, void* d_ws, size_t ws_size,
                              hipStream_t stream) {
    const float* x      = (const float*)d_in[0];
    const float* weight = (const float*)d_in[1];
    const float* w1     = (const float*)d_in[2];
    const float* b1     = (const float*)d_in[3];
    const float* w2     = (const float*)d_in[4];
    const float* b2     = (const float*)d_in[5];
    float* out = (float*)d_out;
    float* agg = (float*)d_ws;   // needs 64*16*64*4 = 256 KB scratch

    moe_attn_agg<<<NB, 256, 0, stream>>>(x, weight, w1, b1, w2, b2, agg);
    moe_conv_wmma<<<NB * NH, 256, 0, stream>>>(x, agg, out);
}